// Pipeline2_62251255989147
// MI455X (gfx1250) — compile-verified
//
#include <hip/hip_runtime.h>
#include <hip/hip_bf16.h>
#include <stdint.h>

typedef float v2f __attribute__((ext_vector_type(2)));
typedef float v8f __attribute__((ext_vector_type(8)));

#define CCH   256
#define HWPX  (512 * 512)
#define NNEG  10
#define NTILES (HWPX / 16)          // 16 pixels per wave-tile
#define KEY_PIX(p) (0x0FFFFFFFu - (unsigned)(p))

__device__ __forceinline__ unsigned orderable_f32(float f) {
    unsigned u = __float_as_uint(f);
    return (u & 0x80000000u) ? ~u : (u | 0x80000000u);
}

// --- kernel 0: reset the global best key (ws is poisoned by the harness) ---
__global__ void k_init(unsigned long long* best) {
    if (threadIdx.x == 0 && blockIdx.x == 0) *best = 0ull;
}

// --- kernel 1: streaming cosine-sim argmax over 256 MB using WMMA ---------
// A (16x4 f32): lane l (h=l>>4, L=l&15): v0 = feats[p0+L][k+2h], v1 = feats[p0+L][k+1+2h]
// Dots: B[kk][n] = emb[k+kk] broadcast across columns -> D[i][*] = dot_i
// Norms: B := A acts as A^T under the f32 16x4 layout -> diag(A*A^T) = ||feat||^2
__global__ __launch_bounds__(256) void k_scan(const float* __restrict__ data,
                                              const float* __restrict__ emb,
                                              unsigned long long* __restrict__ best) {
    const int lane = threadIdx.x & 31;
    const int wid  = threadIdx.x >> 5;
    const int h    = lane >> 4;
    const int L    = lane & 15;
    const int nwaves = (gridDim.x * blockDim.x) >> 5;
    const int gwave  = (blockIdx.x * blockDim.x + threadIdx.x) >> 5;

    const float* ep = emb + 2 * h;      // emb rows for this half-wave's K slots
    unsigned long long bestKey = 0ull;

    for (int tile = gwave; tile < NTILES; tile += nwaves) {
        const int p0 = tile << 4;
        v8f accD = {};                  // dots accumulator (broadcast columns)
        v8f accN = {};                  // Gram accumulator (diagonal = norm^2)
        const float* ap = data + (size_t)(2 * h) * HWPX + p0 + L;

        #pragma unroll 4
        for (int k = 0; k < CCH; k += 4) {
            v2f a; a.x = ap[0];     a.y = ap[HWPX];
            v2f b; b.x = ep[k];     b.y = ep[k + 1];
            accD = __builtin_amdgcn_wmma_f32_16x16x4_f32(
                       false, a, false, b, (short)0, accD, false, false);
            accN = __builtin_amdgcn_wmma_f32_16x16x4_f32(
                       false, a, false, a, (short)0, accN, false, false);
            ap += (size_t)4 * HWPX;
        }

        // Pixel q = r + 8h: dot in accD[r] (all lanes of half h), norm^2 on the
        // Gram diagonal which sits in accN[r] on lanes r (h=0) and r+24 (h=1).
        float dot = 0.f, n2 = -1.f;
        #pragma unroll
        for (int r = 0; r < 8; ++r) {
            bool m = (lane == r) || (lane == r + 24);
            dot = m ? accD[r] : dot;
            n2  = m ? accN[r] : n2;
        }
        const bool valid = (lane < 8) || (lane >= 24);
        const int  pix   = p0 + (lane & 15);          // holds for both halves
        float sim = (valid && n2 > 0.f) ? dot * rsqrtf(n2) : -3.4e38f;

        unsigned long long key =
            ((unsigned long long)orderable_f32(sim) << 32) | KEY_PIX(pix);
        if (key > bestKey) bestKey = key;
    }

    // wave32 max-reduce of packed keys
    #pragma unroll
    for (int off = 16; off > 0; off >>= 1) {
        unsigned long long o = __shfl_xor(bestKey, off, 32);
        if (o > bestKey) bestKey = o;
    }

    __shared__ unsigned long long sbest;
    if (threadIdx.x == 0) sbest = 0ull;
    __syncthreads();
    if (lane == 0) atomicMax(&sbest, bestKey);
    __syncthreads();
    if (threadIdx.x == 0) atomicMax(best, sbest);    // global_atomic_max_u64
}

// --- kernel 2: gather winner + negatives, compute zj = feats[best]*mean(za)
__global__ void k_final(const float* __restrict__ data,
                        const int*   __restrict__ negIdx,
                        const unsigned long long* __restrict__ best,
                        float* __restrict__ out) {
    const int c = threadIdx.x;
    if (c >= CCH) return;
    const int bpix = (int)(0x0FFFFFFFu - (unsigned)(*best & 0xFFFFFFFFull));
    const float* col = data + (size_t)c * HWPX;

    float s = 0.f;
    #pragma unroll
    for (int n = 0; n < NNEG; ++n) {
        float v = col[negIdx[n]];
        out[CCH + n * CCH + c] = v;   // za, row-major [NNEG, C]
        s += v;
    }
    out[c] = col[bpix] * (s * (1.0f / NNEG));   // zj
}

extern "C" void kernel_launch(void* const* d_in, const int* in_sizes, int n_in,
                              void* d_out, int out_size, void* d_ws, size_t ws_size,
                              hipStream_t stream) {
    (void)in_sizes; (void)n_in; (void)out_size; (void)ws_size;
    const float* data   = (const float*)d_in[0];   // [1,256,512,512] f32
    const float* emb    = (const float*)d_in[1];   // [256] f32
    const int*   negIdx = (const int*)d_in[2];     // [10] int32
    float* out = (float*)d_out;                    // zj(256) ++ za(2560)
    unsigned long long* best = (unsigned long long*)d_ws;

    k_init <<<1,    32, 0, stream>>>(best);
    k_scan <<<1024, 256, 0, stream>>>(data, emb, best);
    k_final<<<1,   256, 0, stream>>>(data, negIdx, best, out);
}